// Asterisk_78340203479368
// MI455X (gfx1250) — compile-verified
//
#include <hip/hip_runtime.h>
#include <hip/hip_bf16.h>

// CDNA5 (gfx1250) asterisk-conv: 4 directional 5-tap dilated convs as fp32
// WMMA GEMMs (V_WMMA_F32_16X16X4_F32, exact fp32). NT=64 output columns per
// WG, double-buffered LDS, register-pipelined global->LDS staging so L2
// latency hides behind the 40-WMMA compute block of the previous chunk.

typedef __attribute__((ext_vector_type(2))) float v2f;
typedef __attribute__((ext_vector_type(8))) float v8f;

#define B_DIM   8
#define C_DIM   256
#define H_DIM   128
#define W_DIM   128
#define OUTC    64
#define KTAPS   5
#define HW      (H_DIM * W_DIM)

#define CC      16                 // channels per chunk
#define KC      (CC * KTAPS)       // 80 K values per chunk
#define TSTEPS  (KC / 4)           // 20 WMMA K-steps per chunk
#define NT      64                 // output columns per WG
#define NCHUNK  (C_DIM / CC)       // 16 chunks
#define WS_PITCH 82                // padded weight row (conflict-free, 8B aligned)
#define XS_SIZE (TSTEPS * NT * 4)  // 5120 floats
#define WS_SIZE (OUTC * WS_PITCH)  // 5248 floats
#define XPT     ((KC * NT) / 256)  // 20 x elements staged per thread
#define WPT     ((OUTC * KC) / 256)// 20 w elements staged per thread

__global__ __launch_bounds__(256)
void asterisk_wmma_kernel(const float* __restrict__ x,
                          const float* __restrict__ w_h,  const float* __restrict__ b_h,
                          const float* __restrict__ w_v,  const float* __restrict__ b_v,
                          const float* __restrict__ w_d1, const float* __restrict__ b_d1,
                          const float* __restrict__ w_d2, const float* __restrict__ b_d2,
                          float* __restrict__ out)
{
    __shared__ float xs[2][XS_SIZE];   // B tiles [t][n][k&3], double buffered
    __shared__ float ws[2][WS_SIZE];   // weights [oc][K-chunk], double buffered

    const int w0 = blockIdx.x * NT;    // 0 or 64
    const int h  = blockIdx.y;         // 0..127
    const int bz = blockIdx.z;         // b*4 + branch
    const int b  = bz >> 2;
    const int br = bz & 3;

    // branch tap geometry: offset (mh*d, mw*d), d = 2k-4
    int mh, mw;
    const float* wptr;
    const float* bptr;
    if (br == 0)      { mh = 0;  mw = 1;  wptr = w_h;  bptr = b_h;  }   // horizontal
    else if (br == 1) { mh = 1;  mw = 0;  wptr = w_v;  bptr = b_v;  }   // vertical
    else if (br == 2) { mh = 1;  mw = -1; wptr = w_d1; bptr = b_d1; }   // shear-diag
    else              { mh = -1; mw = 1;  wptr = w_d2; bptr = b_d2; }   // shear-antidiag

    const int tid  = threadIdx.x;
    const int lane = tid & 31;
    const int wave = tid >> 5;         // 0..7
    const int ln   = lane & 15;
    const int g    = lane >> 4;        // half-wave select
    const int ocb  = (wave & 3) * 16;  // oc tile
    const int nb   = (wave >> 2) * 32; // N super-tile (two 16-wide tiles)

    // ---- per-thread chunk-invariant staging descriptors --------------------
    int xg[XPT], xl[XPT];
#pragma unroll
    for (int i = 0; i < XPT; ++i) {
        const int idx = tid + i * 256;
        const int n  = idx & (NT - 1);
        const int r  = idx >> 6;               // r = cc*5 + k
        const int cc = r / KTAPS;
        const int k  = r - cc * KTAPS;
        const int d  = 2 * k - 4;
        const int hh = h + mh * d;
        const int ww = w0 + n + mw * d;
        xg[i] = ((unsigned)hh < (unsigned)H_DIM && (unsigned)ww < (unsigned)W_DIM)
                    ? (cc * HW + hh * W_DIM + ww) : -1;
        xl[i] = (r >> 2) * (NT * 4) + n * 4 + (r & 3);
    }
    int wg[WPT], wl[WPT];
#pragma unroll
    for (int i = 0; i < WPT; ++i) {
        const int idx = tid + i * 256;
        const int oc  = idx / KC;
        const int j   = idx - oc * KC;
        wg[i] = oc * (C_DIM * KTAPS) + j;
        wl[i] = oc * WS_PITCH + j;
    }

    const float* xb = x + (size_t)b * C_DIM * HW;

    float xr[XPT], wr[WPT];
    v8f acc0 = {}, acc1 = {};

    // ---- pipeline stages ---------------------------------------------------
    auto loadRegs = [&](int c0) {
#pragma unroll
        for (int i = 0; i < XPT; ++i)
            xr[i] = (xg[i] >= 0) ? xb[(size_t)c0 * HW + xg[i]] : 0.0f;
#pragma unroll
        for (int i = 0; i < WPT; ++i)
            wr[i] = wptr[(size_t)c0 * KTAPS + wg[i]];
    };
    auto storeLDS = [&](int buf) {
#pragma unroll
        for (int i = 0; i < XPT; ++i) xs[buf][xl[i]] = xr[i];
#pragma unroll
        for (int i = 0; i < WPT; ++i) ws[buf][wl[i]] = wr[i];
    };
    auto compute = [&](int buf) {
        const float* xsb = xs[buf];
        const float* wsb = ws[buf];
#pragma unroll 5
        for (int t = 0; t < TSTEPS; ++t) {
            // A: lane holds M = ocb+ln, K = 4t + 2g + {0,1}; shared by both tiles
            v2f a  = *(const v2f*)&wsb[(ocb + ln) * WS_PITCH + 4 * t + 2 * g];
            v2f b0 = *(const v2f*)&xsb[(t * NT + nb + ln) * 4 + 2 * g];
            v2f b1 = *(const v2f*)&xsb[(t * NT + nb + 16 + ln) * 4 + 2 * g];
            acc0 = __builtin_amdgcn_wmma_f32_16x16x4_f32(false, a, false, b0,
                                                         (short)0, acc0, false, false);
            acc1 = __builtin_amdgcn_wmma_f32_16x16x4_f32(false, a, false, b1,
                                                         (short)0, acc1, false, false);
        }
    };

    // ---- main loop: load(c+1) -> compute(c) -> dump regs -> barrier --------
    loadRegs(0);
    storeLDS(0);
    __syncthreads();
    for (int c = 0; c < NCHUNK; ++c) {
        const int cur = c & 1;
        if (c + 1 < NCHUNK) loadRegs((c + 1) * CC);   // global loads in flight
        compute(cur);                                  // 40 WMMAs hide L2 latency
        if (c + 1 < NCHUNK) storeLDS(cur ^ 1);         // waits land after compute
        __syncthreads();
    }

    // ---- write D tiles + bias: C layout vgpr r -> M = r + 8*g, N = ln ------
    float* outb = out + ((size_t)b * (4 * OUTC) + br * OUTC) * HW + h * W_DIM;
#pragma unroll
    for (int tile = 0; tile < 2; ++tile) {
        const int wcol = w0 + nb + tile * 16 + ln;
        const v8f acc = tile ? acc1 : acc0;
#pragma unroll
        for (int r = 0; r < 8; ++r) {
            const int oc = ocb + r + 8 * g;
            outb[(size_t)oc * HW + wcol] = acc[r] + bptr[oc];
        }
    }
}

extern "C" void kernel_launch(void* const* d_in, const int* in_sizes, int n_in,
                              void* d_out, int out_size, void* d_ws, size_t ws_size,
                              hipStream_t stream) {
    const float* x    = (const float*)d_in[0];
    const float* w_h  = (const float*)d_in[1];
    const float* b_h  = (const float*)d_in[2];
    const float* w_v  = (const float*)d_in[3];
    const float* b_v  = (const float*)d_in[4];
    const float* w_d1 = (const float*)d_in[5];
    const float* b_d1 = (const float*)d_in[6];
    const float* w_d2 = (const float*)d_in[7];
    const float* b_d2 = (const float*)d_in[8];
    float* out = (float*)d_out;

    dim3 grid(W_DIM / NT, H_DIM, B_DIM * 4);   // 2 x 128 x 32 workgroups
    dim3 block(256);                           // 8 wave32 waves
    asterisk_wmma_kernel<<<grid, block, 0, stream>>>(
        x, w_h, b_h, w_v, b_v, w_d1, b_d1, w_d2, b_d2, out);
}